// ReadoutModule_71725953843387
// MI455X (gfx1250) — compile-verified
//
#include <hip/hip_runtime.h>

// ---------------------------------------------------------------------------
// Fused readout kernel for MI455X (gfx1250, wave32, WMMA).
//
// Per row: inv = LN1(s) @ W_inv + b_inv           (64)
//          w   = LN2(s) @ W_w   + b_w             (8192, NEVER materialized)
//          y1  = w1^T @ x1 / 8, y2 = w2^T @ x2 / 8
// Output row = [inv(64) | y1(192) | y2(320)] = 576 f32.
//
// The 16x8192 'w' tile is produced 16x16 at a time with V_WMMA_F32_16X16X4_F32
// and consumed immediately, saving ~2 GB of HBM round-trip per call.
// Each m-block reduction runs as TWO independent 8-deep WMMA chains so the
// XDL pipe always has a second dependency-free stream to interleave.
// ---------------------------------------------------------------------------

typedef __attribute__((ext_vector_type(2))) float v2f;
typedef __attribute__((ext_vector_type(4))) float v4f;
typedef __attribute__((ext_vector_type(8))) float v8f;

#define NROW  30000
#define NSC   64          // scalar features / LN width
#define D1    192         // 64*3
#define D2    320         // 64*5
#define FDIM  576         // NSC + D1 + D2
#define WNUM  8192
#define RPB   16          // rows per block (16-row WMMA tiles)
#define TPAD  66          // LDS row pad: (row*66 + c) % 64 distinct per row

__device__ __forceinline__ v8f wmma_f32_k4(v2f a, v2f b, v8f c) {
  // D(16x16,f32) = A(16x4,f32) x B(4x16,f32) + C
  return __builtin_amdgcn_wmma_f32_16x16x4_f32(
      /*neg_a=*/false, a, /*neg_b=*/false, b,
      /*c_mod=*/(short)0, c, /*reuse_a=*/false, /*reuse_b=*/false);
}

// One wave: for its 16-column k-tile, loop all 64 m-blocks of w{1,2}; build the
// 16x16 w tile with 2x8 chained WMMAs (K = 64 scalars, 4 per step) and contract
// against x (I = 3 for w1, 5 for w2) into per-lane y accumulators.
template <int I>
__device__ __forceinline__ void contract_half(
    const float* __restrict__ Ww, const float* __restrict__ bias_w,
    const float (*Xs)[D1 + D2], const v2f* aw, int kt, int cbase,
    int lane, long grow0, float* __restrict__ out) {
  const int lrow  = lane & 15;          // k column within tile / A row
  const int khalf = (lane >> 4) << 1;   // K sub-offset: 0 (lanes 0-15) or 2
  const int nh    = (lane >> 4) << 3;   // row-half offset for C/D layout
  const int xoff  = (I == 3) ? 0 : D1;  // x1 vs x2 region in LDS
  const int obase = (I == 3) ? NSC : (NSC + D1);

  float y[8 * I] = {};                  // y[j][i], j = local row, k fixed/lane

#pragma unroll 1
  for (int m = 0; m < 64; ++m) {
    const int col = cbase + m * 64 + kt * 16 + lrow;  // column of W_w
    // Two independent accumulator chains (even t / odd t) for XDL ILP.
    v8f acc0 = {};
    v8f acc1 = {};
#pragma unroll
    for (int t = 0; t < 16; t += 2) {
      const int s0 = 4 * t + khalf;     // K index (scalar dim), chain 0
      const int s1 = 4 * (t + 1) + khalf;
      // B fragments (ISA 4x16 f32 layout: v0 holds K={0,2}, v1 holds K={1,3})
      v2f b0, b1;
      b0.x = Ww[(size_t)s0 * WNUM + col];
      b0.y = Ww[(size_t)(s0 + 1) * WNUM + col];
      b1.x = Ww[(size_t)s1 * WNUM + col];
      b1.y = Ww[(size_t)(s1 + 1) * WNUM + col];
      acc0 = wmma_f32_k4(aw[t], b0, acc0);
      acc1 = wmma_f32_k4(aw[t + 1], b1, acc1);
    }
    const float bw = bias_w[col];
    // acc[j]: row n = j + nh, column k = kt*16 + lrow (fixed per lane)
#pragma unroll
    for (int j = 0; j < 8; ++j) {
      const float wv = acc0[j] + acc1[j] + bw;
      const int n = j + nh;
#pragma unroll
      for (int i = 0; i < I; ++i)
        y[j * I + i] += wv * Xs[n][xoff + m * I + i];
    }
  }

  const float nrm = 0.125f;             // 1/sqrt(64)
  const int k = kt * 16 + lrow;
#pragma unroll
  for (int j = 0; j < 8; ++j) {
    const int n = j + nh;
#pragma unroll
    for (int i = 0; i < I; ++i)
      out[(grow0 + n) * FDIM + obase + k * I + i] = y[j * I + i] * nrm;
  }
}

__global__ __launch_bounds__(256) void readout_fused_kernel(
    const float* __restrict__ feat,
    const float* __restrict__ g_inv, const float* __restrict__ be_inv,
    const float* __restrict__ W_inv, const float* __restrict__ bi_inv,
    const float* __restrict__ g_w,  const float* __restrict__ be_w,
    const float* __restrict__ W_w,  const float* __restrict__ bi_w,
    float* __restrict__ out) {
  __shared__ __align__(16) float Tinv[RPB][TPAD];     // LN1 activations
  __shared__ __align__(16) float Tw[RPB][TPAD];       // LN2 activations
  __shared__ __align__(16) float Xs[RPB][D1 + D2];    // x1|x2 per row

  const int tid  = threadIdx.x;
  const int lane = tid & 31;
  const int wave = tid >> 5;                          // 0..7
  const long grow0 = (long)blockIdx.x * RPB;          // first global row

  // ---- Phase 1: LayerNorm of scalars (16 threads per row, shuffle reduce) --
  {
    const int row = tid >> 4;                         // 0..15
    const int sub = tid & 15;                         // 4 scalars each
    const v4f s4 = *(const v4f*)(feat + (grow0 + row) * FDIM + sub * 4);
    float sum = s4.x + s4.y + s4.z + s4.w;
    float sq  = s4.x * s4.x + s4.y * s4.y + s4.z * s4.z + s4.w * s4.w;
#pragma unroll
    for (int m = 8; m >= 1; m >>= 1) {                // stays inside 16-lane row
      sum += __shfl_xor(sum, m, 32);
      sq  += __shfl_xor(sq, m, 32);
    }
    const float mean = sum * (1.0f / 64.0f);
    const float var  = sq * (1.0f / 64.0f) - mean * mean;
    const float rs   = rsqrtf(var + 1e-5f);
#pragma unroll
    for (int j = 0; j < 4; ++j) {
      const int c = sub * 4 + j;
      const float t = (s4[j] - mean) * rs;
      Tinv[row][c] = t * g_inv[c] + be_inv[c];
      Tw[row][c]   = t * g_w[c]  + be_w[c];
    }
  }

  // ---- Phase 2: stage x1|x2 (cols 64..576) into LDS, float4 coalesced ------
#pragma unroll
  for (int it = 0; it < 8; ++it) {
    const int idx = tid + it * 256;                   // 0..2047 float4 chunks
    const int r = idx >> 7;                           // 128 float4 per row
    const int c = idx & 127;
    *(v4f*)(&Xs[r][c * 4]) =
        *(const v4f*)(feat + (grow0 + r) * FDIM + NSC + c * 4);
  }
  __syncthreads();

  // ---- Phase 3: A fragments from LDS (ISA 16x4 f32 A layout) ---------------
  const int lrow  = lane & 15;
  const int khalf = (lane >> 4) << 1;
  v2f aw[16];
#pragma unroll
  for (int t = 0; t < 16; ++t)
    aw[t] = *(const v2f*)(&Tw[lrow][4 * t + khalf]);

  const int kt = wave & 3;

  // ---- Phase 4a: waves 0-3 also produce inv = LN1 @ W_inv + b_inv ----------
  if (wave < 4) {
    v2f ai[16];
#pragma unroll
    for (int t = 0; t < 16; ++t)
      ai[t] = *(const v2f*)(&Tinv[lrow][4 * t + khalf]);
    const int col = kt * 16 + lrow;
    v8f acc0 = {};
    v8f acc1 = {};
#pragma unroll
    for (int t = 0; t < 16; t += 2) {
      const int s0 = 4 * t + khalf;
      const int s1 = 4 * (t + 1) + khalf;
      v2f b0, b1;
      b0.x = W_inv[s0 * NSC + col];
      b0.y = W_inv[(s0 + 1) * NSC + col];
      b1.x = W_inv[s1 * NSC + col];
      b1.y = W_inv[(s1 + 1) * NSC + col];
      acc0 = wmma_f32_k4(ai[t], b0, acc0);
      acc1 = wmma_f32_k4(ai[t + 1], b1, acc1);
    }
    const float bi = bi_inv[col];
    const int nh = (lane >> 4) << 3;
#pragma unroll
    for (int j = 0; j < 8; ++j)
      out[(grow0 + j + nh) * FDIM + col] = acc0[j] + acc1[j] + bi;
  }

  // ---- Phase 4b: dominant fused w-matmul + contraction ---------------------
  if (wave < 4)
    contract_half<3>(W_w, bi_w, Xs, aw, kt, /*cbase=*/0,    lane, grow0, out);
  else
    contract_half<5>(W_w, bi_w, Xs, aw, kt, /*cbase=*/4096, lane, grow0, out);
}

extern "C" void kernel_launch(void* const* d_in, const int* in_sizes, int n_in,
                              void* d_out, int out_size, void* d_ws, size_t ws_size,
                              hipStream_t stream) {
  (void)in_sizes; (void)n_in; (void)out_size; (void)d_ws; (void)ws_size;
  const float* feat   = (const float*)d_in[0];
  const float* g_inv  = (const float*)d_in[1];
  const float* be_inv = (const float*)d_in[2];
  const float* W_inv  = (const float*)d_in[3];
  const float* bi_inv = (const float*)d_in[4];
  const float* g_w    = (const float*)d_in[5];
  const float* be_w   = (const float*)d_in[6];
  const float* W_w    = (const float*)d_in[7];
  const float* bi_w   = (const float*)d_in[8];
  float* out = (float*)d_out;

  dim3 grid(NROW / RPB);   // 1875 blocks, exact (30000 % 16 == 0)
  dim3 block(256);         // 8 wave32 waves
  readout_fused_kernel<<<grid, block, 0, stream>>>(
      feat, g_inv, be_inv, W_inv, bi_inv, g_w, be_w, W_w, bi_w, out);
}